// SpatialProbLoss_52510270161367
// MI455X (gfx1250) — compile-verified
//
#include <hip/hip_runtime.h>
#include <stdint.h>

typedef float v2f __attribute__((ext_vector_type(2)));
typedef float v8f __attribute__((ext_vector_type(8)));

#define N_PTS   8192
#define CAP     32
#define OCC_THR 0.5f

// ---------------- workspace layout ----------------
// [0, 32KB)            : uint32 cnt[8192]
// [32KB, 32KB + 2MB)   : uint64 cand[8192 * 32]   key = (f32bits(d)<<32) | j

__global__ void SPL_zero_counts(uint32_t* __restrict__ cnt) {
  int i = blockIdx.x * blockDim.x + threadIdx.x;
  if (i < N_PTS) cnt[i] = 0u;
}

// ---------------- Phase 1: WMMA candidate generation ----------------
// One wave per block. Block = (rowTile, chunk): 16 true rows x 256 pred cols.
// d^2 = |t|^2 + |p|^2 - 2*(t.p); cross term via V_WMMA_F32_16X16X4_F32.
// Pred-tile loads are software-pipelined one tile ahead so the global load
// latency overlaps the WMMA + filter work of the previous tile.
__global__ __launch_bounds__(32)
void SPL_candgen(const float4* __restrict__ pred, const float4* __restrict__ truc,
                 uint32_t* __restrict__ cnt, unsigned long long* __restrict__ cand) {
  __shared__ float4 T[16];
  __shared__ float  Tn2[16];
  const int lane  = threadIdx.x;
  const int i0    = blockIdx.x * 16;      // base true row
  const int chunk = blockIdx.y;           // 16 pred tiles each

  if (lane < 16) {
    float4 t = truc[i0 + lane];
    T[lane]   = t;
    Tn2[lane] = t.x * t.x + t.y * t.y + t.z * t.z;
  }
  __syncthreads();

  // A fragment: 16x4 f32 (K = x,y,z,0). VGPR0 = K0|K2, VGPR1 = K1|K3.
  const int  half = lane >> 4;            // 0: lanes 0-15, 1: lanes 16-31
  float4 tA = T[lane & 15];
  v2f a;
  a[0] = half ? tA.z : tA.x;
  a[1] = half ? 0.0f : tA.y;

  const int n      = lane & 15;           // N index of this lane (C/D layout)
  const int mbase  = half * 8;            // M = mbase + v
  const int jbase  = chunk * 256;         // first pred index of this chunk

  // pipeline prologue: load tile 0
  float4 p4 = pred[jbase + n];

#pragma unroll 2
  for (int t = 0; t < 16; ++t) {
    // prefetch next tile (clamped to offset 0 on the last tile; value unused)
    const int tn = (t < 15) ? (t + 1) : 0;
    float4 p4n = pred[jbase + tn * 16 + n];

    const int j0 = jbase + t * 16;

    // B fragment: 4x16 f32, mirrored layout: VGPR0 = K0|K2, VGPR1 = K1|K3.
    v2f b;
    b[0] = half ? p4.z : p4.x;
    b[1] = half ? 0.0f : p4.y;

    v8f c = {};
    // dot(t_M, p_N) for the 16x16 tile on the matrix pipe
    c = __builtin_amdgcn_wmma_f32_16x16x4_f32(false, a, false, b,
                                              (short)0, c, false, false);

    if (p4.w >= OCC_THR) {
      float pn2 = p4.x * p4.x + p4.y * p4.y + p4.z * p4.z;
#pragma unroll
      for (int v = 0; v < 8; ++v) {
        const int M = mbase + v;
        float d2a = Tn2[M] + pn2 - 2.0f * c[v];
        if (d2a <= 1.02f) {                         // slack pre-filter
          float4 tr = T[M];
          if (tr.w >= OCC_THR) {
            // exact reference-order recompute (no fma contraction)
            float dx = __fsub_rn(tr.x, p4.x);
            float dy = __fsub_rn(tr.y, p4.y);
            float dz = __fsub_rn(tr.z, p4.z);
            float d2 = __fadd_rn(__fadd_rn(__fmul_rn(dx, dx), __fmul_rn(dy, dy)),
                                 __fmul_rn(dz, dz));
            float d = sqrtf(d2);
            if (d <= 1.0f) {
              int i = i0 + M;
              uint32_t slot = atomicAdd(&cnt[i], 1u);
              if (slot < CAP) {
                unsigned long long key =
                    (((unsigned long long)__float_as_uint(d)) << 32) |
                    (unsigned long long)(uint32_t)(j0 + n);
                cand[(size_t)i * CAP + slot] = key;
              }
            }
          }
        }
      }
    }
    p4 = p4n;
  }
}

// ---------------- Phase 2: sequential greedy matching (1 wave) ----------------
__global__ __launch_bounds__(32)
void SPL_match(const float4* __restrict__ pred, const float4* __restrict__ truc,
               const uint32_t* __restrict__ cnt,
               const unsigned long long* __restrict__ cand,
               float* __restrict__ out) {
  __shared__ uint32_t      taken[N_PTS / 32];   // 1 KB bitmask over pred cols
  __shared__ unsigned char scnt[N_PTS];         // 8 KB clamped counts
  __shared__ float         tp[N_PTS];           // 32 KB true_p cache
  const int lane = threadIdx.x;

  for (int k = lane; k < N_PTS / 32; k += 32) taken[k] = 0u;
  float occ_local = 0.0f;
  for (int k = lane; k < N_PTS; k += 32) {
    uint32_t c = cnt[k];
    scnt[k] = (unsigned char)(c < CAP ? c : CAP);
    float w = truc[k].w;
    tp[k] = w;
    if (w >= OCC_THR) occ_local += 1.0f;
  }
  // wave32 reduce of occ_local
  for (int off = 16; off > 0; off >>= 1) occ_local += __shfl_xor(occ_local, off, 32);
  __syncthreads();

  const unsigned long long INVALID = ~0ull;

  // software pipeline: prefetch row i+1's candidates (+ their pred_p) during row i
  unsigned long long key_next = INVALID;
  float pw_next = 0.0f;
  {
    uint32_t c0 = scnt[0];
    if (lane < (int)c0) {
      key_next = cand[lane];
      pw_next  = pred[(uint32_t)key_next].w;
    }
  }

  float sum_d = 0.0f, sum_p = 0.0f, n_matched = 0.0f;

  for (int i = 0; i < N_PTS; ++i) {
    unsigned long long key = key_next;
    float pw = pw_next;

    // prefetch next row
    key_next = INVALID;
    pw_next  = 0.0f;
    if (i + 1 < N_PTS) {
      uint32_t cn = scnt[i + 1];
      if (lane < (int)cn) {
        key_next = cand[(size_t)(i + 1) * CAP + lane];
        pw_next  = pred[(uint32_t)key_next].w;
      }
    }

    // mask out already-taken columns
    if (key != INVALID) {
      uint32_t j = (uint32_t)key;
      if (taken[j >> 5] & (1u << (j & 31))) key = INVALID;
    }

    // wave32 min-reduce of (key, pw) -> argmin with reference tie-break
    unsigned long long m = key;
    float mpw = pw;
    for (int off = 16; off > 0; off >>= 1) {
      unsigned lo = (unsigned)m, hi = (unsigned)(m >> 32);
      unsigned olo = __shfl_xor(lo, off, 32);
      unsigned ohi = __shfl_xor(hi, off, 32);
      float    opw = __shfl_xor(mpw, off, 32);
      unsigned long long o = (((unsigned long long)ohi) << 32) | olo;
      if (o < m) { m = o; mpw = opw; }
    }

    if (lane == 0 && m != INVALID) {
      uint32_t j = (uint32_t)m;
      taken[j >> 5] |= (1u << (j & 31));
      float d  = __uint_as_float((uint32_t)(m >> 32));
      float dp = tp[i] - mpw;
      sum_d     += d;
      sum_p     += dp * dp;
      n_matched += 1.0f;
    }
    __syncthreads();  // taken update visible before next row's mask test
  }

  if (lane == 0) {
    float mean_dist = sum_d / n_matched;
    float prob_mse  = sum_p / n_matched;
    float unmatched = occ_local - n_matched;
    // spatial_error + prob_error = mean_dist + 10*unm + prob_mse + unm
    out[0] = mean_dist + prob_mse + 11.0f * unmatched;
  }
}

extern "C" void kernel_launch(void* const* d_in, const int* in_sizes, int n_in,
                              void* d_out, int out_size, void* d_ws, size_t ws_size,
                              hipStream_t stream) {
  (void)in_sizes; (void)n_in; (void)out_size; (void)ws_size;
  const float4* pred = (const float4*)d_in[0];   // pred_cloud (8192,4) f32
  const float4* truc = (const float4*)d_in[1];   // true_cloud (8192,4) f32

  uint32_t* cnt = (uint32_t*)d_ws;
  unsigned long long* cand =
      (unsigned long long*)((char*)d_ws + (size_t)N_PTS * sizeof(uint32_t));

  SPL_zero_counts<<<N_PTS / 256, 256, 0, stream>>>(cnt);

  dim3 grid(N_PTS / 16, 32);   // 512 row tiles x 32 chunks of 16 pred tiles
  SPL_candgen<<<grid, 32, 0, stream>>>(pred, truc, cnt, cand);

  SPL_match<<<1, 32, 0, stream>>>(pred, truc, cnt, cand, (float*)d_out);
}